// OrthogonalButterfly_3143916060795
// MI455X (gfx1250) — compile-verified
//
#include <hip/hip_runtime.h>

#define WIDTH   1024
#define HALF    512
#define DEPTH   10
#define RPB     8          // rows per block
#define THREADS 256
#define NV4     ((RPB * WIDTH / 4) / THREADS)   // float4 chunks per thread = 8

typedef float f4  __attribute__((ext_vector_type(4)));
typedef int   v4i __attribute__((vector_size(16)));          // matches builtin's V4i
typedef __attribute__((address_space(1))) v4i* gv4i_p;       // global  v4i*
typedef __attribute__((address_space(3))) v4i* lv4i_p;       // LDS     v4i*

#if defined(__HIP_DEVICE_COMPILE__) && \
    __has_builtin(__builtin_amdgcn_global_load_async_to_lds_b128) && \
    __has_builtin(__builtin_amdgcn_global_store_async_from_lds_b128)
#define HAVE_ASYNC_LDS 1
#else
#define HAVE_ASYNC_LDS 0
#endif

__device__ __forceinline__ void wait_async0() {
#if __has_builtin(__builtin_amdgcn_s_wait_asynccnt)
    __builtin_amdgcn_s_wait_asynccnt(0);
#else
    asm volatile("s_wait_asynccnt 0" ::: "memory");
#endif
}

__device__ __forceinline__ int rotl10(int x, int s) {
    // 10-bit rotate-left; x < 1024, 0 <= s <= 9
    return ((x << s) | (x >> (10 - s))) & 1023;
}

__global__ __launch_bounds__(THREADS)
void butterfly_kernel(const float* __restrict__ X,
                      const float* __restrict__ params,
                      float* __restrict__ Y) {
    __shared__ __align__(16) float A[RPB * WIDTH];   // 32 KB

    const int    t    = threadIdx.x;
    const size_t base = (size_t)blockIdx.x * (RPB * WIDTH);

    // ---------- stream 8 rows (32 KB) into LDS ----------
#if HAVE_ASYNC_LDS
    {
        const f4* g = (const f4*)(X + base);
        f4*       l = (f4*)A;
#pragma unroll
        for (int k = 0; k < NV4; ++k) {
            const int idx = t + THREADS * k;
            __builtin_amdgcn_global_load_async_to_lds_b128(
                (gv4i_p)(g + idx), (lv4i_p)(l + idx), 0, 0);
        }
        wait_async0();
    }
#else
    {
        const f4* g = (const f4*)(X + base);
        f4*       l = (f4*)A;
#pragma unroll
        for (int k = 0; k < NV4; ++k) {
            const int idx = t + THREADS * k;
            l[idx] = g[idx];
        }
    }
#endif
    __syncthreads();

    // ---------- in-place radix-4 butterfly: 5 double-stages ----------
#pragma unroll
    for (int d2 = 0; d2 < DEPTH / 2; ++d2) {
        const int s0 = 2 * d2;
        const int s1 = s0 + 1;
        const int m1 = 1 << (9 - s0);   // stage s0 partner mask
        const int m2 = m1 >> 1;         // stage s1 partner mask
        // insert two zero bits at positions (8-s0),(9-s0): quad base slot
        const int p  = ((t & ~(m2 - 1)) << 2) | (t & (m2 - 1));

        const int ja = rotl10(p,      s0);
        const int jb = rotl10(p + m2, s0);
        const int jc = rotl10(p,      s1);
        const int jd = rotl10(p + m1, s1);

        float ca, sa, cb, sb, cc, sc, cd, sd;
        __sincosf(params[ja * DEPTH + s0], &sa, &ca);
        __sincosf(params[jb * DEPTH + s0], &sb, &cb);
        __sincosf(params[jc * DEPTH + s1], &sc, &cc);
        __sincosf(params[jd * DEPTH + s1], &sd, &cd);

#pragma unroll
        for (int r = 0; r < RPB; ++r) {
            float* Ar = A + r * WIDTH;
            const float x0 = Ar[p];
            const float x1 = Ar[p + m1];
            const float x2 = Ar[p + m2];
            const float x3 = Ar[p + m1 + m2];
            // stage s0 (pairs across m1)
            const float y0 = fmaf(ca, x0,  sa * x1);
            const float y1 = fmaf(ca, x1, -sa * x0);
            const float y2 = fmaf(cb, x2,  sb * x3);
            const float y3 = fmaf(cb, x3, -sb * x2);
            // stage s1 (pairs across m2), written back in place
            Ar[p]           = fmaf(cc, y0,  sc * y2);
            Ar[p + m2]      = fmaf(cc, y2, -sc * y0);
            Ar[p + m1]      = fmaf(cd, y1,  sd * y3);
            Ar[p + m1 + m2] = fmaf(cd, y3, -sd * y1);
        }
        __syncthreads();
    }

    // ---------- stream 8 rows back out ----------
#if HAVE_ASYNC_LDS
    {
        f4* g = (f4*)(Y + base);
        f4* l = (f4*)A;
#pragma unroll
        for (int k = 0; k < NV4; ++k) {
            const int idx = t + THREADS * k;
            __builtin_amdgcn_global_store_async_from_lds_b128(
                (gv4i_p)(g + idx), (lv4i_p)(l + idx), 0, 0);
        }
        wait_async0();
    }
#else
    {
        f4* g = (f4*)(Y + base);
        f4* l = (f4*)A;
#pragma unroll
        for (int k = 0; k < NV4; ++k) {
            const int idx = t + THREADS * k;
            g[idx] = l[idx];
        }
    }
#endif
}

extern "C" void kernel_launch(void* const* d_in, const int* in_sizes, int n_in,
                              void* d_out, int out_size, void* d_ws, size_t ws_size,
                              hipStream_t stream) {
    const float* X      = (const float*)d_in[0];   // (32768, 1024) fp32
    const float* params = (const float*)d_in[1];   // (512, 10) fp32
    float*       Y      = (float*)d_out;           // (32768, 1024) fp32

    const int batch  = in_sizes[0] / WIDTH;        // 32768
    const int blocks = batch / RPB;                // 4096

    butterfly_kernel<<<dim3(blocks), dim3(THREADS), 0, stream>>>(X, params, Y);
}